// NegativeLogLikelihood_38474317038009
// MI455X (gfx1250) — compile-verified
//
#include <hip/hip_runtime.h>

// ---------------------------------------------------------------------------
// Cox negative log partial likelihood, N = 262144 samples, D = 128 columns.
//
// Memory-bound column-wise scan:  HBM floor = 128MiB / 23.3 TB/s ~ 5.8us.
// Two-pass chunked scan; pass 2 re-reads pred which fits in the 192MB L2,
// so total HBM traffic stays ~one pass.  Tiles are staged to LDS with the
// CDNA5 async global->LDS B128 path (ASYNCcnt) when available.
// ---------------------------------------------------------------------------

#define N_COLS          128
#define CHUNK_ROWS      128                       // rows per block (== blockDim)
#define TILE_ROWS       32                        // rows staged per LDS tile
#define TILES_PER_CHUNK (CHUNK_ROWS / TILE_ROWS)  // 4
#define TILE_FLOATS     (TILE_ROWS * N_COLS)      // 4096 floats = 16 KB

typedef int v4i __attribute__((vector_size(4 * sizeof(int))));
typedef v4i __attribute__((address_space(1)))* gv4i_p;  // global int4*
typedef v4i __attribute__((address_space(3)))* lv4i_p;  // LDS int4*

// Copy one 32x128 fp32 tile (16 KB) global -> LDS; 128 threads x 8 B128 ops.
// Ends with a workgroup barrier so the tile is visible to every wave.
__device__ __forceinline__ void load_tile(const float* __restrict__ g,
                                          float* l, int tid) {
#if defined(__gfx1250__) && __has_builtin(__builtin_amdgcn_global_load_async_to_lds_b128)
#pragma unroll
  for (int k = 0; k < 8; ++k) {
    const int off = (tid + k * N_COLS) * 4;  // float index of this float4
    __builtin_amdgcn_global_load_async_to_lds_b128(
        (gv4i_p)(g + off), (lv4i_p)(l + off),
        /*imm offset*/ 0, /*cpol*/ 0);
  }
#if __has_builtin(__builtin_amdgcn_s_wait_asynccnt)
  __builtin_amdgcn_s_wait_asynccnt(0);
#else
  asm volatile("s_wait_asynccnt 0" ::: "memory");
#endif
#else
  // Fallback: synchronous 128-bit copies.
#pragma unroll
  for (int k = 0; k < 8; ++k) {
    const int off = tid + k * N_COLS;
    reinterpret_cast<float4*>(l)[off] =
        reinterpret_cast<const float4*>(g)[off];
  }
#endif
  __syncthreads();
}

// Pass 1: per-chunk column sums of exp(pred); exact per-chunk yevent counts.
__global__ void cox_pass1(const float* __restrict__ pred,
                          const float* __restrict__ yevent,
                          float* __restrict__ chunkSum,
                          float* __restrict__ yevPartial) {
  __shared__ __align__(16) float tile[TILE_FLOATS];
  __shared__ float red[N_COLS];
  const int  c    = blockIdx.x;
  const int  t    = threadIdx.x;          // column this thread owns
  const long base = (long)c * CHUNK_ROWS; // first row of this chunk

  float s = 0.f;
  for (int tl = 0; tl < TILES_PER_CHUNK; ++tl) {
    load_tile(pred + (base + (long)tl * TILE_ROWS) * N_COLS, tile, t);
#pragma unroll
    for (int r = 0; r < TILE_ROWS; ++r)
      s += __expf(tile[r * N_COLS + t]);
    __syncthreads();  // tile consumed before next async overwrite
  }
  chunkSum[(long)c * N_COLS + t] = s;

  // yevent in {0,1}: integer-valued sums are exact in fp32 in any order.
  red[t] = yevent[base + t];              // CHUNK_ROWS == N_COLS == 128
  __syncthreads();
  if (t == 0) {
    float ys = 0.f;
    for (int i = 0; i < N_COLS; ++i) ys += red[i];
    yevPartial[c] = ys;
  }
}

// Pass 2 setup: per-column exclusive scan of chunk sums (f64 accumulate),
// plus n_observed. One block of 128 threads; coalesced row-wise access.
__global__ void cox_scan(const float* __restrict__ chunkSum,
                         const float* __restrict__ yevPartial,
                         float* __restrict__ prefix,
                         float* __restrict__ nObsOut,
                         int nChunks) {
  const int t = threadIdx.x;
  double run = 0.0;
  for (int c = 0; c < nChunks; ++c) {
    prefix[(long)c * N_COLS + t] = (float)run;
    run += (double)chunkSum[(long)c * N_COLS + t];
  }
  if (t == 0) {
    float ys = 0.f;
    for (int c = 0; c < nChunks; ++c) ys += yevPartial[c];
    nObsOut[0] = ys;
  }
}

// Pass 2: reconstruct running risk sum, accumulate yevent*(x - log(run)).
// pred is L2-resident from pass 1 (128 MiB < 192 MB L2).
__global__ void cox_pass2(const float* __restrict__ pred,
                          const float* __restrict__ yevent,
                          const float* __restrict__ prefix,
                          float* __restrict__ partial) {
  __shared__ __align__(16) float tile[TILE_FLOATS];
  __shared__ float yev[CHUNK_ROWS];
  const int  c    = blockIdx.x;
  const int  t    = threadIdx.x;
  const long base = (long)c * CHUNK_ROWS;

  yev[t] = yevent[base + t];  // visible after the barrier inside load_tile
  float run = prefix[(long)c * N_COLS + t];
  float acc = 0.f;

  for (int tl = 0; tl < TILES_PER_CHUNK; ++tl) {
    load_tile(pred + (base + (long)tl * TILE_ROWS) * N_COLS, tile, t);
#pragma unroll
    for (int r = 0; r < TILE_ROWS; ++r) {
      const float x = tile[r * N_COLS + t];
      run += __expf(x);                       // inclusive risk-set sum
      const float ye = yev[tl * TILE_ROWS + r];
      if (ye != 0.f)                          // wave-uniform branch (~30% skip)
        acc += ye * (x - __logf(run));
    }
    __syncthreads();
  }
  partial[(long)c * N_COLS + t] = acc;
}

// Final: reduce chunk partials per column, normalize, negate.
__global__ void cox_final(const float* __restrict__ partial,
                          const float* __restrict__ nObs,
                          float* __restrict__ out, int nChunks) {
  const int t = threadIdx.x;
  double s = 0.0;
  for (int c = 0; c < nChunks; ++c)
    s += (double)partial[(long)c * N_COLS + t];
  out[t] = (float)(-(s / (double)nObs[0]));
}

extern "C" void kernel_launch(void* const* d_in, const int* in_sizes, int n_in,
                              void* d_out, int out_size, void* d_ws, size_t ws_size,
                              hipStream_t stream) {
  const float* pred   = (const float*)d_in[0];
  // d_in[1] = ytime: sorted-order assumption only; unused by the math.
  const float* yevent = (const float*)d_in[2];
  float*       out    = (float*)d_out;

  const int n       = in_sizes[0] / N_COLS;   // 262144
  const int nChunks = n / CHUNK_ROWS;         // 2048 (N divides exactly)

  // Workspace layout (floats): ~3.01 MB total.
  float* ws         = (float*)d_ws;
  float* chunkSum   = ws;                                   // nChunks*128
  float* prefix     = ws + (size_t)nChunks * N_COLS;        // nChunks*128
  float* partial    = ws + 2 * (size_t)nChunks * N_COLS;    // nChunks*128
  float* yevPartial = ws + 3 * (size_t)nChunks * N_COLS;    // nChunks
  float* nObs       = yevPartial + nChunks;                 // 1

  cox_pass1<<<nChunks, N_COLS, 0, stream>>>(pred, yevent, chunkSum, yevPartial);
  cox_scan <<<1,       N_COLS, 0, stream>>>(chunkSum, yevPartial, prefix, nObs, nChunks);
  cox_pass2<<<nChunks, N_COLS, 0, stream>>>(pred, yevent, prefix, partial);
  cox_final<<<1,       N_COLS, 0, stream>>>(partial, nObs, out, nChunks);
}